// GQNN_29025388986760
// MI455X (gfx1250) — compile-verified
//
#include <hip/hip_runtime.h>
#include <hip/hip_bf16.h>

typedef __attribute__((ext_vector_type(2))) float v2f;
typedef __attribute__((ext_vector_type(8))) float v8f;

#define N_NODES 20000
#define N_EDGES 640000
#define DIM 128

// ---------------------------------------------------------------------------
// zero scratch (agg + cnt are contiguous)
__global__ void sage_zero(float* __restrict__ p, int n) {
    int i = blockIdx.x * blockDim.x + threadIdx.x;
    if (i < n) p[i] = 0.0f;
}

// ---------------------------------------------------------------------------
// edge scatter: one wave32 per edge, lane handles 4 features (float4 load,
// 4 float atomics). cnt incremented once per edge by lane 0.
__global__ void sage_scatter(const float* __restrict__ feat,
                             const long long* __restrict__ src,
                             const long long* __restrict__ dst,
                             float* __restrict__ agg,
                             float* __restrict__ cnt,
                             int nEdges) {
    int gid  = blockIdx.x * blockDim.x + threadIdx.x;
    int edge = gid >> 5;
    int lane = gid & 31;
    if (edge >= nEdges) return;
    int s = (int)src[edge];
    int d = (int)dst[edge];
    const float4 v = *(const float4*)(feat + (size_t)s * DIM + lane * 4);
    float* a = agg + (size_t)d * DIM + lane * 4;
    atomicAdd(a + 0, v.x);
    atomicAdd(a + 1, v.y);
    atomicAdd(a + 2, v.z);
    atomicAdd(a + 3, v.w);
    if (lane == 0) atomicAdd(&cnt[d], 1.0f);
}

// ---------------------------------------------------------------------------
// in-place mean: agg[i][f] /= max(cnt[i], 1)
__global__ void sage_mean(float* __restrict__ agg, const float* __restrict__ cnt,
                          int total) {
    int i = blockIdx.x * blockDim.x + threadIdx.x;
    if (i < total) {
        float c = cnt[i >> 7];
        agg[i] = agg[i] / fmaxf(c, 1.0f);
    }
}

// ---------------------------------------------------------------------------
// Fused SAGE linear: out = relu(A0 @ Wl^T + bias + A1 @ Wr^T)
// A0/A1: [nRows, 128] f32, Wl/Wr: [128, 128] row-major (out_feat, in_feat).
// One wave per 16x16 output tile; K marched in steps of 4 with
// V_WMMA_F32_16X16X4_F32 (full fp32 precision).
//
// f32 16x4 A layout : lanes 0-15 -> M=lane, {K, K+1}; lanes 16-31 -> {K+2, K+3}
// f32 4x16 B layout : lanes 0-15 -> N=lane, rows {K, K+1}; lanes 16-31 rows {K+2,K+3}
// f32 16x16 C/D     : lanes 0-15 -> N=lane, VGPR r = M=r; lanes 16-31 -> M=8+r
__global__ void __launch_bounds__(256)
sage_gemm_wmma(const float* __restrict__ A0, const float* __restrict__ A1,
               const float* __restrict__ Wl, const float* __restrict__ Wr,
               const float* __restrict__ bias, float* __restrict__ out,
               int nRows) {
    const int lane  = threadIdx.x & 31;
    const int wave  = threadIdx.x >> 5;
    const int tile  = blockIdx.x * (blockDim.x >> 5) + wave;
    const int nTilesN = DIM / 16;                 // 8
    const int m0 = (tile / nTilesN) * 16;
    const int n0 = (tile % nTilesN) * 16;
    if (m0 >= nRows) return;                      // uniform per wave (nRows%16==0)

    const int lhalf = lane >> 4;                  // 0 or 1
    const int lmod  = lane & 15;
    const size_t arow = (size_t)(m0 + lmod) * DIM;
    const size_t brow = (size_t)(n0 + lmod) * DIM;

    v8f acc = {};
    #pragma unroll 4
    for (int k = 0; k < DIM; k += 4) {
        const int kk = k + 2 * lhalf;             // this lane's K pair
        v2f a0 = *(const v2f*)(A0 + arow + kk);
        v2f b0 = *(const v2f*)(Wl + brow + kk);   // B[k][n] = Wl[n][k]
        acc = __builtin_amdgcn_wmma_f32_16x16x4_f32(
            false, a0, false, b0, (short)0, acc, false, false);
        v2f a1 = *(const v2f*)(A1 + arow + kk);
        v2f b1 = *(const v2f*)(Wr + brow + kk);
        acc = __builtin_amdgcn_wmma_f32_16x16x4_f32(
            false, a1, false, b1, (short)0, acc, false, false);
    }

    const float b = bias[n0 + lmod];
    const int rbase = m0 + 8 * lhalf;
    #pragma unroll
    for (int r = 0; r < 8; ++r) {
        float v = acc[r] + b;
        v = v > 0.0f ? v : 0.0f;
        out[(size_t)(rbase + r) * DIM + (n0 + lmod)] = v;
    }
}

// ---------------------------------------------------------------------------
// heads: one wave per node. preds = h.wp + bp; diffs = sigmoid(h.wd + bd)
// out[0:N] = preds - diffs ; out[N:2N] = preds + diffs
__global__ void sage_heads(const float* __restrict__ h,
                           const float* __restrict__ wp, const float* __restrict__ bp,
                           const float* __restrict__ wd, const float* __restrict__ bd,
                           float* __restrict__ out, int n) {
    int gid  = blockIdx.x * blockDim.x + threadIdx.x;
    int node = gid >> 5;
    int lane = gid & 31;
    if (node >= n) return;
    const float4 hv = *(const float4*)(h + (size_t)node * DIM + lane * 4);
    const float4 pv = *(const float4*)(wp + lane * 4);
    const float4 dv = *(const float4*)(wd + lane * 4);
    float p = hv.x * pv.x + hv.y * pv.y + hv.z * pv.z + hv.w * pv.w;
    float d = hv.x * dv.x + hv.y * dv.y + hv.z * dv.z + hv.w * dv.w;
    #pragma unroll
    for (int off = 16; off > 0; off >>= 1) {
        p += __shfl_down(p, off, 32);
        d += __shfl_down(d, off, 32);
    }
    if (lane == 0) {
        p += bp[0];
        d += bd[0];
        float sg = 1.0f / (1.0f + __expf(-d));
        out[node]     = p - sg;
        out[n + node] = p + sg;
    }
}

// ---------------------------------------------------------------------------
extern "C" void kernel_launch(void* const* d_in, const int* in_sizes, int n_in,
                              void* d_out, int out_size, void* d_ws, size_t ws_size,
                              hipStream_t stream) {
    const float*     x   = (const float*)d_in[0];
    const long long* ei  = (const long long*)d_in[1];   // int64 [2, E]
    const float*     w1l = (const float*)d_in[2];
    const float*     b1l = (const float*)d_in[3];
    const float*     w1r = (const float*)d_in[4];
    const float*     w2l = (const float*)d_in[5];
    const float*     b2l = (const float*)d_in[6];
    const float*     w2r = (const float*)d_in[7];
    const float*     wp  = (const float*)d_in[8];
    const float*     bp  = (const float*)d_in[9];
    const float*     wd  = (const float*)d_in[10];
    const float*     bd  = (const float*)d_in[11];
    float*           out = (float*)d_out;

    const int N = N_NODES, E = N_EDGES;
    float* ws  = (float*)d_ws;
    float* agg = ws;                        // [N,128] -> becomes mean in place
    float* cnt = agg + (size_t)N * DIM;     // [N]
    float* h1  = cnt + N;                   // [N,128]
    float* h2  = h1 + (size_t)N * DIM;      // [N,128]

    const long long* src = ei;
    const long long* dst = ei + E;

    const int zn = N * DIM + N;             // agg + cnt contiguous
    const int featTot = N * DIM;

    // ---- layer 1 ----
    sage_zero<<<(zn + 255) / 256, 256, 0, stream>>>(agg, zn);
    sage_scatter<<<(E * 32 + 255) / 256, 256, 0, stream>>>(x, src, dst, agg, cnt, E);
    sage_mean<<<(featTot + 255) / 256, 256, 0, stream>>>(agg, cnt, featTot);
    sage_gemm_wmma<<<(N / 16) * (DIM / 16) / 8, 256, 0, stream>>>(
        agg, x, w1l, w1r, b1l, h1, N);

    // ---- layer 2 ----
    sage_zero<<<(zn + 255) / 256, 256, 0, stream>>>(agg, zn);
    sage_scatter<<<(E * 32 + 255) / 256, 256, 0, stream>>>(h1, src, dst, agg, cnt, E);
    sage_mean<<<(featTot + 255) / 256, 256, 0, stream>>>(agg, cnt, featTot);
    sage_gemm_wmma<<<(N / 16) * (DIM / 16) / 8, 256, 0, stream>>>(
        agg, h1, w2l, w2r, b2l, h2, N);

    // ---- heads ----
    sage_heads<<<(N * 32 + 255) / 256, 256, 0, stream>>>(h2, wp, bp, wd, bd, out, N);
}